// ConvCapsDAE_5832565588546
// MI455X (gfx1250) — compile-verified
//
#include <hip/hip_runtime.h>
#include <hip/hip_bf16.h>

typedef __attribute__((ext_vector_type(2))) float v2f;
typedef __attribute__((ext_vector_type(8))) float v8f;

// Problem constants (match reference)
#define BATCH   4
#define BC      32      // B_ capsule types
#define KK      3       // kernel
#define CC      32      // C_ input caps
#define WSP     14      // spatial in
#define HH      16      // 4x4 pose
#define NKL     (BC*KK*KK)          // 288
#define SPOS    (WSP*WSP)           // 196
#define CSP     (CC*SPOS)           // 6272
#define OUT_HW  29
#define NCOLS   (BATCH*SPOS*4)      // 3136 columns: (b, s, w)
#define NTILES  (NCOLS/16)          // 196 column tiles
#define VWS_OFF (BATCH*CSP*HH)      // 401408 floats of sd, then v_ws

// ---------------- Pass 0: sd = sqrt(var) ----------------
__global__ void sqrt_var_kernel(const float* __restrict__ var,
                                float* __restrict__ sd, int n) {
    int i = blockIdx.x * blockDim.x + threadIdx.x;
    if (i < n) sd[i] = sqrtf(var[i]);
}

// ---------------- Pass 1: fused votes + einsum via f32 WMMA ----------------
// Grid (288, 2), 128 threads (4 waves). Workgroup = (nkl, tile-half).
// Wave-global id wg = blockIdx.y*4 + wave owns column tiles t = wg, wg+8, ...
// Per tile: noise chunk (32c x 4s x 64B = 8KB) is double-buffered in LDS via
// global_load_async_to_lds_b128 (ASYNCcnt); mu/sd come from L2 in groups of 8
// K-steps; contraction is V_WMMA_F32_16X16X4_F32 over K=128 (one c per step).
__global__ __launch_bounds__(128)
void votes_gemm_kernel(const float* __restrict__ poses,   // mu: [b][c*196+s][hh]
                       const float* __restrict__ Wt,      // [nkl][c][u][v]
                       const float* __restrict__ noise,   // [b][nkl][c*196+s][hh]
                       const float* __restrict__ sd,      // same layout as mu
                       float* __restrict__ vws) {         // [b][nkl][s][u*4+w]
    __shared__ float nbuf[4][2][2048];                    // 4 waves x 2 bufs x 8KB

    const int nkl  = blockIdx.x;
    const int lane = threadIdx.x & 31;
    const int wave = threadIdx.x >> 5;
    const int wg   = blockIdx.y * 4 + wave;   // 0..7, tile stride 8
    const int m    = lane & 15;               // A-row (u) / B-column within tile
    const int h    = lane >> 4;               // lane half selects K pair {2h, 2h+1}
    const int sl   = m >> 2;                  // s within tile (0..3)
    const int w    = m & 3;
    const int hh0  = (2 * h) * 4 + w;         // v = 2h
    const int hh1  = hh0 + 4;                 // v = 2h+1

    // Preload A fragments for the full K-loop: Wt[nkl, c, u, v], rows u<4 valid.
    const float* Wb = Wt + (size_t)nkl * (CC * HH);
    v2f afrag[CC];
    #pragma unroll
    for (int c = 0; c < CC; ++c) {
        v2f a; a.x = 0.f; a.y = 0.f;
        if (m < 4) {
            const float* p = Wb + c * 16 + m * 4 + 2 * h;  // v = 2h, 2h+1
            a.x = p[0]; a.y = p[1];
        }
        afrag[c] = a;
    }

    const size_t noise_nkl = (size_t)nkl * (size_t)(CSP * HH);

    // Async copy of tile t's noise chunk into nbuf[wave][pbuf].
    // Copy-lane role: row c = lane. Per row: 4 s-values x 64B = 256 contiguous
    // bytes in global; 16 x B128 with shared immediate offsets 0..240 (the ISA
    // adds INST_OFFSET to both the LDS and global addresses).
    auto copy_tile = [&](int t, int pbuf) {
        const int col0 = t * 16;
        const int bb   = col0 / (SPOS * 4);
        const int s0   = (col0 - bb * (SPOS * 4)) >> 2;
        const float* gsrc = noise + ((size_t)bb * NKL) * (size_t)(CSP * HH) + noise_nkl
                          + ((size_t)lane * SPOS + s0) * HH;
        unsigned lds = (unsigned)(uintptr_t)(&nbuf[wave][pbuf][lane * 64]);
        #pragma unroll
        for (int k2 = 0; k2 < 16; ++k2) {
            asm volatile("global_load_async_to_lds_b128 %0, %1, off offset:%c2"
                         :: "v"(lds), "v"(gsrc), "i"(k2 * 16) : "memory");
        }
    };

    copy_tile(wg, 0);     // prologue prefetch (every wg < NTILES)

    int p = 0;
    for (int t = wg; t < NTILES; t += 8, p ^= 1) {
        const int tn = t + 8;
        if (tn < NTILES) {
            copy_tile(tn, 1 - p);                                // prefetch next
            asm volatile("s_wait_asynccnt 16" ::: "memory");     // tile t done
        } else {
            asm volatile("s_wait_asynccnt 0" ::: "memory");
        }

        const int col = t * 16 + m;            // global column: (b, s, w)
        const int b   = col / (SPOS * 4);      // tiles never straddle b (784%16==0)
        const int s   = (col - b * (SPOS * 4)) >> 2;

        const float* mu_b = poses + (size_t)b * CSP * HH;
        const float* sd_b = sd    + (size_t)b * CSP * HH;
        const float* lbuf = &nbuf[wave][p][0];

        v8f acc = {};
        #pragma unroll
        for (int g = 0; g < 4; ++g) {          // 4 groups of 8 K-steps
            float mb0[8], mb1[8], sb0[8], sb1[8];
            #pragma unroll
            for (int q = 0; q < 8; ++q) {      // batch L2 loads for MLP
                const int c = g * 8 + q;
                const size_t off = (size_t)(c * SPOS + s) * HH;
                mb0[q] = mu_b[off + hh0]; mb1[q] = mu_b[off + hh1];
                sb0[q] = sd_b[off + hh0]; sb1[q] = sd_b[off + hh1];
            }
            #pragma unroll
            for (int q = 0; q < 8; ++q) {
                const int c = g * 8 + q;
                const float n0 = lbuf[c * 64 + sl * 16 + hh0];
                const float n1 = lbuf[c * 64 + sl * 16 + hh1];
                v2f bfrag;
                bfrag.x = fmaf(sb0[q], n0, mb0[q]);
                bfrag.y = fmaf(sb1[q], n1, mb1[q]);
                acc = __builtin_amdgcn_wmma_f32_16x16x4_f32(
                    false, afrag[c], false, bfrag, (short)0, acc, false, false);
            }
        }

        // D layout: lane L<16 holds rows M=0..7 in vgpr 0..7; only u=0..3 valid.
        if (h == 0) {
            float* o = vws + (((size_t)b * NKL + nkl) * SPOS + s) * HH + w;
            o[0]  = acc[0];
            o[4]  = acc[1];
            o[8]  = acc[2];
            o[12] = acc[3];
        }
    }
}

// ---------------- Pass 2: transposed-conv gather into recon ----------------
// recon[b,n,r,c,hh] = sum over (k,i): 2i+k==r, (l,j): 2j+l==c of
//                     vws[b, (n*3+k)*3+l, j*14+i, hh]
__global__ void recon_kernel(const float* __restrict__ vws,
                             float* __restrict__ out, int total) {
    int idx = blockIdx.x * blockDim.x + threadIdx.x;
    if (idx >= total) return;
    int hh = idx & 15;
    int t  = idx >> 4;
    int cc = t % OUT_HW; t /= OUT_HW;
    int rr = t % OUT_HW; t /= OUT_HW;
    int n  = t % BC;
    int b  = t / BC;

    float acc = 0.f;
    #pragma unroll
    for (int k = 0; k < KK; ++k) {
        int ri = rr - k;
        if (ri < 0 || (ri & 1) || ri > 26) continue;
        int i = ri >> 1;
        #pragma unroll
        for (int l = 0; l < KK; ++l) {
            int cj = cc - l;
            if (cj < 0 || (cj & 1) || cj > 26) continue;
            int j = cj >> 1;
            int nkl = (n * KK + k) * KK + l;
            acc += vws[(((size_t)b * NKL + nkl) * SPOS + (j * WSP + i)) * HH + hh];
        }
    }
    out[idx] = acc;
}

extern "C" void kernel_launch(void* const* d_in, const int* in_sizes, int n_in,
                              void* d_out, int out_size, void* d_ws, size_t ws_size,
                              hipStream_t stream) {
    (void)in_sizes; (void)n_in; (void)ws_size;
    const float* poses = (const float*)d_in[0];
    const float* var   = (const float*)d_in[1];
    const float* Wt    = (const float*)d_in[2];
    const float* noise = (const float*)d_in[3];
    float* out = (float*)d_out;

    float* sd  = (float*)d_ws;             // 401,408 floats
    float* vws = sd + VWS_OFF;             // 3,612,672 floats

    // Pass 0: sd = sqrt(var)
    {
        int n = VWS_OFF;
        sqrt_var_kernel<<<(n + 255) / 256, 256, 0, stream>>>(var, sd, n);
    }
    // Pass 1: fused votes + per-(n,k,l) GEMM (WMMA f32 16x16x4, async LDS stream)
    {
        dim3 grid(NKL, 2);
        votes_gemm_kernel<<<grid, 128, 0, stream>>>(poses, Wt, noise, sd, vws);
    }
    // Pass 2: gather into recon
    recon_kernel<<<(out_size + 255) / 256, 256, 0, stream>>>(vws, out, out_size);
}